// Pool_80135499809385
// MI455X (gfx1250) — compile-verified
//
#include <hip/hip_runtime.h>
#include <hip/hip_bf16.h>
#include <math.h>

// Problem constants (match reference)
#define BB    4
#define NN    4096
#define DD    512
#define KSEL  2048

typedef float v2f __attribute__((ext_vector_type(2)));
typedef float v8f __attribute__((ext_vector_type(8)));

// ---------------------------------------------------------------------------
// CDNA5 async global->LDS helpers (ASYNCcnt-tracked, no VGPR staging).
// lds_off: byte offset within the workgroup LDS allocation (low 32 bits of the
// generic pointer per ISA 10.2).
// ---------------------------------------------------------------------------
__device__ __forceinline__ void async_load_b128_to_lds(unsigned lds_off,
                                                       const void* gaddr)
{
    asm volatile("global_load_async_to_lds_b128 %0, %1, off"
                 :: "v"(lds_off), "v"(gaddr)
                 : "memory");
}

__device__ __forceinline__ void wait_async0()
{
#if __has_builtin(__builtin_amdgcn_s_wait_asynccnt)
    __builtin_amdgcn_s_wait_asynccnt(0);
#else
    asm volatile("s_wait_asynccnt 0x0" ::: "memory");
#endif
}

__device__ __forceinline__ float fast_sigmoid(float x)
{
#if __has_builtin(__builtin_amdgcn_rcpf)
    return __builtin_amdgcn_rcpf(1.0f + __expf(-x));
#else
    return 1.0f / (1.0f + __expf(-x));
#endif
}

// ---------------------------------------------------------------------------
// Kernel 1: scores[b,n] = sigmoid(sum_d h[b,n,d] * sf[b,d])
// One wave computes 16 rows via V_WMMA_F32_16X16X4_F32 over D in K=4 steps.
// A (16x4 f32): lanes 0-15 = rows M=0..15 holding K=0 (v0), K=1 (v1);
// lanes 16-31 hold K=2 (v0), K=3 (v1). B is the section feature replicated
// across all 16 N-columns, so every output column equals the score vector.
// C/D: VGPR i -> M=i (lanes 0-15) / M=8+i (lanes 16-31); column 0 lives in
// lane 0 (M=0..7) and lane 16 (M=8..15).
// ---------------------------------------------------------------------------
__global__ __launch_bounds__(32) void scores_kernel(
    const float* __restrict__ h, const float* __restrict__ sf,
    float* __restrict__ scores)
{
    const int tile = blockIdx.x;              // 0 .. BB*(NN/16)-1
    const int b    = tile / (NN / 16);
    const int m0   = (tile % (NN / 16)) * 16;
    const int lane = threadIdx.x;             // wave32
    const int row  = m0 + (lane & 15);
    const int koff = (lane >> 4) * 2;         // 0 or 2

    const float* __restrict__ hrow = h  + ((size_t)b * NN + row) * DD;
    const float* __restrict__ sfb  = sf + (size_t)b * DD;

    v8f c = {};                               // zero accumulator
    #pragma unroll 4
    for (int k = 0; k < DD; k += 4) {
        v2f a, bm;
        a.x  = hrow[k + koff];
        a.y  = hrow[k + koff + 1];
        bm.x = sfb[k + koff];                 // replicated across N columns
        bm.y = sfb[k + koff + 1];
        c = __builtin_amdgcn_wmma_f32_16x16x4_f32(
                /*neg_a=*/false, a, /*neg_b=*/false, bm,
                /*c_mod=*/(short)0, c, /*reuse_a=*/false, /*reuse_b=*/false);
    }

    if ((lane & 15) == 0) {                   // lane 0 -> M=0..7, lane 16 -> M=8..15
        const int mbase = m0 + (lane >> 4) * 8;
        #pragma unroll
        for (int i = 0; i < 8; ++i)
            scores[(size_t)b * NN + mbase + i] = fast_sigmoid(c[i]);
    }
}

// ---------------------------------------------------------------------------
// Kernel 2: exact top-K with sorted (descending) output, per batch.
// One workgroup per batch; bitonic sort of 4096 (score, idx) pairs in LDS
// (32 KB of the 320 KB WGP LDS). Scores staged via async-to-LDS B128.
// Descending by score, ascending index on ties (matches jax.lax.top_k).
// ---------------------------------------------------------------------------
__global__ __launch_bounds__(1024) void topk_kernel(
    const float* __restrict__ scores,
    int* __restrict__ idx_out, float* __restrict__ val_out)
{
    __shared__ float s_val[NN];
    __shared__ int   s_idx[NN];

    const int b   = blockIdx.x;
    const int tid = threadIdx.x;

    // Async-stage the 16 KB score row: 1024 lanes x one B128 each.
    {
        const float4* __restrict__ ssrc = (const float4*)(scores + (size_t)b * NN);
        const unsigned vbase = (unsigned)(uintptr_t)&s_val[0];
        async_load_b128_to_lds(vbase + (unsigned)tid * 16u, ssrc + tid);
    }
    for (int i = tid; i < NN; i += 1024)
        s_idx[i] = i;
    wait_async0();
    __syncthreads();

    for (int k = 2; k <= NN; k <<= 1) {
        for (int j = k >> 1; j > 0; j >>= 1) {
            for (int i = tid; i < NN; i += 1024) {
                const int ixj = i ^ j;
                if (ixj > i) {
                    const bool desc = ((i & k) == 0);   // descending segment
                    float v1 = s_val[i],  v2 = s_val[ixj];
                    int   i1 = s_idx[i],  i2 = s_idx[ixj];
                    // out-of-order for descending: smaller value, or tie w/ larger idx
                    const bool bad = (v1 < v2) || (v1 == v2 && i1 > i2);
                    if (bad == desc) {
                        s_val[i] = v2; s_val[ixj] = v1;
                        s_idx[i] = i2; s_idx[ixj] = i1;
                    }
                }
            }
            __syncthreads();
        }
    }

    for (int i = tid; i < KSEL; i += 1024) {
        idx_out[b * KSEL + i] = s_idx[i];
        val_out[b * KSEL + i] = s_val[i];
    }
}

// ---------------------------------------------------------------------------
// Kernel 3: row gathers (fully coalesced float4 / B128 traffic).
//   g_mask[b,i,:] = g3[b, idx[b,i], :]              (4096 floats per row)
//   new_h [b,i,:] = h [b, idx[b,i], :] * vals[b,i]  ( 512 floats per row)
// One block per selected row. Direct B128 load/store is the optimal data
// path here (no reuse, pure streaming), so no LDS staging.
// ---------------------------------------------------------------------------
__global__ __launch_bounds__(256) void rowgather_kernel(
    const float* __restrict__ g3, const float* __restrict__ h,
    const int* __restrict__ idx, const float* __restrict__ vals,
    float* __restrict__ g_mask, float* __restrict__ new_h)
{
    const int gid = blockIdx.x;               // b*KSEL + i
    const int b   = gid / KSEL;
    const int r   = idx[gid];
    const float v = vals[gid];

    const float4* __restrict__ src = (const float4*)(g3 + ((size_t)b * NN + r) * NN);
    float4* __restrict__ dst       = (float4*)(g_mask + (size_t)gid * NN);
    __builtin_prefetch(src, 0, 1);            // global_prefetch_b8
    #pragma unroll 2
    for (int t = threadIdx.x; t < NN / 4; t += 256)
        dst[t] = src[t];

    const float4* __restrict__ hsrc = (const float4*)(h + ((size_t)b * NN + r) * DD);
    float4* __restrict__ hdst       = (float4*)(new_h + (size_t)gid * DD);
    for (int t = threadIdx.x; t < DD / 4; t += 256) {
        float4 x = hsrc[t];
        x.x *= v; x.y *= v; x.z *= v; x.w *= v;
        hdst[t] = x;
    }
}

// ---------------------------------------------------------------------------
// Kernel 4: row+column gathers.
//   g_section [b,i,j] = g1[b, idx[b,i], idx[b,j]]
//   g_sentence[b,i,j] = g2[b, idx[b,i], idx[b,j]]
// One block per output row; the 2048 column indices are async-staged once
// into LDS (8 KB, 512 B128 transfers) while the source rows are prefetched;
// the random part is then only 4B gathers inside one cached 16 KB row.
// ---------------------------------------------------------------------------
__global__ __launch_bounds__(256) void colgather_kernel(
    const float* __restrict__ g1, const float* __restrict__ g2,
    const int* __restrict__ idx,
    float* __restrict__ g_section, float* __restrict__ g_sentence)
{
    __shared__ int s_cidx[KSEL];

    const int gid = blockIdx.x;               // b*KSEL + i
    const int b   = gid / KSEL;

    // Async-stage the index row: 512 x B128 across 256 lanes (2 each).
    {
        const int4* __restrict__ isrc = (const int4*)(idx + b * KSEL);
        const unsigned ibase = (unsigned)(uintptr_t)&s_cidx[0];
        #pragma unroll
        for (int it = 0; it < 2; ++it) {
            const int e = threadIdx.x + it * 256;   // B128 element, 0..511
            async_load_b128_to_lds(ibase + (unsigned)e * 16u, isrc + e);
        }
    }

    const int r = idx[gid];
    const float* __restrict__ row1 = g1 + ((size_t)b * NN + r) * NN;
    const float* __restrict__ row2 = g2 + ((size_t)b * NN + r) * NN;
    __builtin_prefetch(row1, 0, 1);
    __builtin_prefetch(row2, 0, 1);

    wait_async0();
    __syncthreads();

    float* __restrict__ o1 = g_section  + (size_t)gid * KSEL;
    float* __restrict__ o2 = g_sentence + (size_t)gid * KSEL;

    #pragma unroll 4
    for (int t = threadIdx.x; t < KSEL; t += 256) {
        const int c = s_cidx[t];
        o1[t] = row1[c];
        o2[t] = row2[c];
    }
}

// ---------------------------------------------------------------------------
// Launch
// ---------------------------------------------------------------------------
extern "C" void kernel_launch(void* const* d_in, const int* in_sizes, int n_in,
                              void* d_out, int out_size, void* d_ws, size_t ws_size,
                              hipStream_t stream)
{
    const float* g1 = (const float*)d_in[0];   // [B,N,N]
    const float* g2 = (const float*)d_in[1];   // [B,N,N]
    const float* g3 = (const float*)d_in[2];   // [B,N,N]
    const float* h  = (const float*)d_in[3];   // [B,N,D]
    const float* sf = (const float*)d_in[4];   // [B,1,D]

    // Outputs concatenated flat in return order.
    float* g_section  = (float*)d_out;                                   // B*K*K
    float* g_sentence = g_section  + (size_t)BB * KSEL * KSEL;           // B*K*K
    float* g_mask     = g_sentence + (size_t)BB * KSEL * KSEL;           // B*K*N
    float* new_h      = g_mask     + (size_t)BB * KSEL * NN;             // B*K*D

    // Workspace: scores (B*N f32), idx (B*K i32), vals (B*K f32) = 128 KB.
    float* scores = (float*)d_ws;
    int*   idx    = (int*)(scores + (size_t)BB * NN);
    float* vals   = (float*)(idx + (size_t)BB * KSEL);

    scores_kernel   <<<BB * (NN / 16), 32,   0, stream>>>(h, sf, scores);
    topk_kernel     <<<BB,             1024, 0, stream>>>(scores, idx, vals);
    rowgather_kernel<<<BB * KSEL,      256,  0, stream>>>(g3, h, idx, vals, g_mask, new_h);
    colgather_kernel<<<BB * KSEL,      256,  0, stream>>>(g1, g2, idx, g_section, g_sentence);
}